// PrimalDual_89223650607191
// MI455X (gfx1250) — compile-verified
//
#include <hip/hip_runtime.h>
#include <hip/hip_bf16.h>
#include <math.h>

// ---------------------------------------------------------------------------
// Primal-dual multi-label segmentation solver, CDNA5 (gfx1250) version.
//   H=W=256, l=16 labels, proj=136, repeats=10 (constants from setup_inputs).
//   musum = mux @ M via V_WMMA_F32_16X16X4_F32, A-tiles staged into LDS by the
//   Tensor Data Mover (tensor_load_to_lds + s_wait_tensorcnt), B-operand built
//   branchlessly from a packed (k1,k2) LDS table.
//   t = cs[k2+1]-cs[k1] via per-pixel LDS cumsum (cheaper than a 2nd GEMM).
// Memory-bound: ~430 MB streamed per iteration at 23.3 TB/s -> WMMA/TDM hide
// the gather latency; layouts: pixel-space [pix][z], proj-space [d][p][pix].
// ---------------------------------------------------------------------------

typedef float v2f __attribute__((ext_vector_type(2)));
typedef float v8f __attribute__((ext_vector_type(8)));
typedef unsigned int u32x4 __attribute__((ext_vector_type(4)));
typedef unsigned int u32x8 __attribute__((ext_vector_type(8)));

constexpr int   HDIM  = 256;
constexpr int   WDIM  = 256;
constexpr int   NPIX  = HDIM * WDIM;        // 65536
constexpr int   LLAB  = 16;
constexpr int   PROJ  = 136;                // l*(l-1)/2 + l
constexpr int   PXL   = NPIX * LLAB;        // 1048576
constexpr int   PROJN = PROJ * NPIX;        // per-dim projection plane
constexpr int   PROJ2 = 2 * PROJN;          // both dims

constexpr float HS     = 256.0f;            // fwd/bwd diff scale (= H)
constexpr float LS     = 16.0f;             // label diff scale (= l)
constexpr float SIGMAP = (1.0f / 256.0f) / 19.0f;   // res/(3+l)
constexpr float TAUU   = (1.0f / 256.0f) / 6.0f;    // res/6
constexpr float TAU    = 1.0f / 36.0f;              // 1/(2+proj/4)
constexpr float LMBDA  = 0.5f;
constexpr float NU     = 0.1f;
constexpr float NUH    = 25.6f;                     // nu*H

// ws float layout: u | ubar | px[2] | pt | mux | mubarx | sx | nrj
constexpr size_t OFF_U    = 0;
constexpr size_t OFF_UB   = PXL;
constexpr size_t OFF_PX   = 2 * (size_t)PXL;
constexpr size_t OFF_PT   = 4 * (size_t)PXL;
constexpr size_t OFF_MUX  = 5 * (size_t)PXL;
constexpr size_t OFF_NRJ  = OFF_MUX + 3 * (size_t)PROJ2;

// ---------------------------------------------------------------------------
// TDM: async 2D tile load (16 floats/row x 136 rows, row stride NPIX floats)
// into LDS. D# groups per CDNA5 ISA (2-group / 2D form).
// ---------------------------------------------------------------------------
__device__ __forceinline__ void tdm_load_tile_f32(unsigned lds_byte_addr,
                                                  const float* gptr,
                                                  unsigned tile_d0,   // elems/row
                                                  unsigned tile_d1,   // rows
                                                  unsigned row_stride) // elems
{
    unsigned long long ga = (unsigned long long)(const void*)gptr;
    u32x4 g0;
    g0[0] = 1u;                                               // count=1 (valid user D#)
    g0[1] = lds_byte_addr;                                    // lds_addr
    g0[2] = (unsigned)(ga & 0xffffffffu);                     // global_addr[31:0]
    g0[3] = (unsigned)((ga >> 32) & 0x01ffffffu) | (2u << 30);// addr[56:32] | type=2
    u32x8 g1;
    g1[0] = (2u << 16);                                       // data_size=4B, no flags
    g1[1] = ((unsigned)NPIX & 0xffffu) << 16;                 // tensor_dim0[15:0]
    g1[2] = ((unsigned)NPIX >> 16) | (tile_d1 << 16);         // dim0[31:16] | tensor_dim1[15:0]
    g1[3] = (tile_d0 << 16);                                  // tensor_dim1[31:16]=0 | tile_dim0
    g1[4] = tile_d1;                                          // tile_dim1 | tile_dim2=0
    g1[5] = row_stride;                                       // tensor_dim0_stride[31:0]
    g1[6] = 0u;                                               // stride[47:32] | dim1_stride lo
    g1[7] = 0u;                                               // dim1_stride hi
    asm volatile("tensor_load_to_lds %0, %1" :: "s"(g0), "s"(g1) : "memory");
}

// ---------------------------------------------------------------------------
__global__ __launch_bounds__(256) void pd_init_u(const float* __restrict__ f,
                                                 float* __restrict__ ws) {
    int tid = blockIdx.x * 256 + threadIdx.x;     // grid covers PXL exactly
    float v = f[tid >> 4];                        // broadcast f across labels
    ws[OFF_U  + tid] = v;
    ws[OFF_UB + tid] = v;
}

__global__ __launch_bounds__(256) void pd_zero(float* __restrict__ ws) {
    size_t n = 3 * (size_t)PXL + 3 * (size_t)PROJ2 + 1;
    float* base = ws + OFF_PX;
    for (size_t i = (size_t)blockIdx.x * 256 + threadIdx.x; i < n;
         i += (size_t)gridDim.x * 256)
        base[i] = 0.0f;
}

// ---------------------------------------------------------------------------
// Dual update: one wave owns a 16-pixel x 16-label tile.  128 thr = 4 waves.
//   stage 0: TDM loads both A tiles (mux[d], 136x16 f32) into LDS (async)
//   stage 1: musum[d] = mux[d] @ M : 34 K-steps, 2 interleaved WMMA accums
//   stage 2: cubic-root dual projection -> new px, pt  (elementwise, C-layout)
//   stage 3: sx / mux / mubarx streaming update using LDS cumsum for t
// ---------------------------------------------------------------------------
__global__ __launch_bounds__(128) void pd_dual(const float* __restrict__ f,
                                               float* __restrict__ ws) {
    float* ubar = ws + OFF_UB;
    float* px   = ws + OFF_PX;
    float* pt   = ws + OFF_PT;
    float* mux  = ws + OFF_MUX;
    float* mubx = mux + PROJ2;
    float* sx   = mux + 2 * (size_t)PROJ2;

    __shared__ int   skp[PROJ];              // packed k1 | (k2<<8)
    __shared__ float sA  [4][2][PROJ][16];   // TDM-staged A tiles (68 KB)
    __shared__ float sraw[4][2][16][16];
    __shared__ float scs [4][2][16][17];     // stride 17 -> bank-conflict free

    const int tid  = threadIdx.x;
    const int wv   = tid >> 5;
    const int lane = tid & 31;
    const int hi   = lane >> 4;
    const int lo   = lane & 15;

    // wave-uniform copies for the TDM descriptor (SGPR operands)
    const int wu  = __builtin_amdgcn_readfirstlane(tid) >> 5;
    const int pbu = ((int)blockIdx.x * 4 + wu) * 16;          // tile base pixel
    const int pb  = pbu;

    // ---- stage 0: issue async TDM tile loads for both dims
    #pragma unroll
    for (int d = 0; d < 2; ++d) {
        unsigned lds_addr = (unsigned)(size_t)&sA[wu][d][0][0];
        tdm_load_tile_f32(lds_addr, mux + (size_t)d * PROJN + pbu,
                          16u, (unsigned)PROJ, (unsigned)NPIX);
    }

    // combo table p -> (a,b), packed (blockDim 128 < 136 -> strided fill)
    for (int p = tid; p < PROJ; p += 128) {
        int a = 0, rem = p;
        while (rem >= (LLAB - a)) { rem -= (LLAB - a); ++a; }
        skp[p] = a | ((a + rem) << 8);
    }
    __syncthreads();                          // skp visible
    __builtin_amdgcn_s_wait_tensorcnt(0);     // sA[wv] tiles landed (per-wave)

    // ---- stage 1: musum via WMMA, both dims share the B operand
    const float* sA0 = &sA[wv][0][0][0];
    const float* sA1 = &sA[wv][1][0][0];
    v8f acc0 = {0.f, 0.f, 0.f, 0.f, 0.f, 0.f, 0.f, 0.f};
    v8f acc1 = {0.f, 0.f, 0.f, 0.f, 0.f, 0.f, 0.f, 0.f};
    #pragma unroll 4
    for (int k0 = 0; k0 < PROJ; k0 += 4) {
        int ka = k0 + hi * 2;                 // this half-wave's K pair
        v2f A0, A1, B;
        A0.x = sA0[(ka + 0) * 16 + lo];
        A0.y = sA0[(ka + 1) * 16 + lo];
        A1.x = sA1[(ka + 0) * 16 + lo];
        A1.y = sA1[(ka + 1) * 16 + lo];
        int pka = skp[ka], pkb = skp[ka + 1];
        B.x = (float)(((pka & 255) <= lo) & (lo <= (pka >> 8)));   // branchless
        B.y = (float)(((pkb & 255) <= lo) & (lo <= (pkb >> 8)));
        acc0 = __builtin_amdgcn_wmma_f32_16x16x4_f32(
                   false, A0, false, B, (short)0, acc0, false, false);
        acc1 = __builtin_amdgcn_wmma_f32_16x16x4_f32(
                   false, A1, false, B, (short)0, acc1, false, false);
    }
    v8f mus[2] = {acc0, acc1};

    // ---- stage 2: elementwise dual projection in WMMA C-layout
    #pragma unroll
    for (int r = 0; r < 8; ++r) {
        int q   = pb + r + hi * 8;            // pixel
        int z   = lo;                         // label
        int idx = q * LLAB + z;
        int x   = q & (WDIM - 1);
        int y   = q >> 8;

        float uc = ubar[idx];
        float d0 = (y < HDIM - 1) ? (ubar[idx + WDIM * LLAB] - uc) * HS : 0.0f;
        float d1 = (x < WDIM - 1) ? (ubar[idx + LLAB] - uc) * HS : 0.0f;
        float dz = (z < LLAB - 1) ? (ubar[idx + 1] - uc) * LS : 0.0f;

        float p0o = px[idx];
        float p1o = px[PXL + idx];
        float pto = pt[idx];

        float ux0 = p0o + SIGMAP * (d0 + mus[0][r]);
        float ux1 = p1o + SIGMAP * (d1 + mus[1][r]);
        float ut  = pto + SIGMAP * dz;

        float fv  = f[q];
        float klf = (float)(z + 1) * (1.0f / 16.0f) - fv;
        float pen = LMBDA * klf * klf;

        float n2  = ux0 * ux0 + ux1 * ux1;
        float Bv  = 0.25f * n2 - pen;
        bool  mk  = ut < Bv;
        float yv  = ut + pen;
        float nrm = sqrtf(n2);
        float av  = 0.5f * nrm;
        float bb  = (2.0f / 3.0f) * (1.0f - 0.5f * yv);
        float sb  = sqrtf(fmaxf(-bb, 0.0f));
        float sb3 = sb * sb * sb;
        float dv  = (bb < 0.0f) ? (av - sb3) * (av + sb3) : av * av + bb * bb * bb;
        float cv  = cbrtf(av + sqrtf(fmaxf(dv, 0.0f)));
        bool  m1  = (dv >= 0.0f) && (cv == 0.0f);
        bool  m3  = dv < 0.0f;
        float sb3s = (sb > 0.0f) ? sb3 : 1.0f;
        float rat  = fminf(fmaxf(av / sb3s, -1.0f), 1.0f);
        float csafe = (cv == 0.0f) ? 1.0f : cv;
        float vv = m1 ? 0.0f
                      : (m3 ? 2.0f * sb * cosf(acosf(rat) * (1.0f / 3.0f))
                            : cv - bb / csafe);
        float nsafe = (nrm == 0.0f) ? 1.0f : nrm;
        float scl   = (nrm == 0.0f) ? 0.0f : 2.0f * vv / nsafe;

        float pn0 = mk ? scl * ux0 : ux0;
        float pn1 = mk ? scl * ux1 : ux1;
        float ptn = mk ? 0.25f * (pn0 * pn0 + pn1 * pn1) - pen : ut;

        px[idx]       = pn0;
        px[PXL + idx] = pn1;
        pt[idx]       = ptn;
        sraw[wv][0][r + hi * 8][z] = pn0;
        sraw[wv][1][r + hi * 8][z] = pn1;
    }
    __syncthreads();

    // ---- per-pixel exclusive cumsum over labels (lane = (dim, pixel))
    {
        int d = hi, pl = lo;
        float acc = 0.0f;
        scs[wv][d][pl][0] = 0.0f;
        #pragma unroll
        for (int z = 0; z < LLAB; ++z) {
            acc += sraw[wv][d][pl][z];
            scs[wv][d][pl][z + 1] = acc;
        }
    }
    __syncthreads();

    // ---- stage 3: streaming projection-space update (lanes: 2 p x 16 pix)
    for (int j = 0; j < PROJ / 2; ++j) {
        int p   = j * 2 + hi;
        int e   = p * NPIX + pb + lo;
        int pk  = skp[p];
        int k1p = pk & 255, k2p = pk >> 8;

        __builtin_prefetch(&sx[e + 2 * NPIX], 0, 1);   // global_prefetch_b8

        float mx0 = sx[e]         - mubx[e];
        float mx1 = sx[PROJN + e] - mubx[PROJN + e];
        float sn  = sqrtf(mx0 * mx0 + mx1 * mx1);
        float sc  = (sn > NUH) ? (NU / sn) : 1.0f;
        float sx0 = mx0 * sc, sx1 = mx1 * sc;

        float t0 = scs[wv][0][lo][k2p + 1] - scs[wv][0][lo][k1p];
        float t1 = scs[wv][1][lo][k2p + 1] - scs[wv][1][lo][k1p];

        float m0o = mux[e],         m1o = mux[PROJN + e];
        float m0n = m0o + TAU * (sx0 - t0);
        float m1n = m1o + TAU * (sx1 - t1);

        sx[e]           = sx0;          sx[PROJN + e]   = sx1;
        mux[e]          = m0n;          mux[PROJN + e]  = m1n;
        mubx[e]         = 2.0f * m0n - m0o;
        mubx[PROJN + e] = 2.0f * m1n - m1o;
    }
}

// ---------------------------------------------------------------------------
// Primal update: un = clip(u + tauu*(div px + div_t pt)), boundary labels fixed.
// ---------------------------------------------------------------------------
__global__ __launch_bounds__(256) void pd_primal(float* __restrict__ ws,
                                                 int computeNrj) {
    float* u    = ws + OFF_U;
    float* ubar = ws + OFF_UB;
    float* px   = ws + OFF_PX;
    float* pt   = ws + OFF_PT;
    float* nrj  = ws + OFF_NRJ;

    int tid = blockIdx.x * 256 + threadIdx.x;      // grid covers PXL exactly
    int q = tid >> 4, z = tid & 15;
    int x = q & (WDIM - 1), y = q >> 8;

    float a0  = (y < HDIM - 1) ? px[tid] : 0.0f;
    float b0  = (y > 0) ? px[tid - WDIM * LLAB] : 0.0f;
    float dx0 = (a0 - b0) * HS;

    float a1  = (x < WDIM - 1) ? px[PXL + tid] : 0.0f;
    float b1  = (x > 0) ? px[PXL + tid - LLAB] : 0.0f;
    float dx1 = (a1 - b1) * HS;

    float at = (z < LLAB - 1) ? pt[tid] : 0.0f;
    float bt = (z > 0) ? pt[tid - 1] : 0.0f;
    float dt = (at - bt) * LS;

    float uo = u[tid];
    float Dv = uo + TAUU * (dx0 + dx1 + dt);
    float un = fminf(fmaxf(Dv, 0.0f), 1.0f);
    if (z == 0)        un = 1.0f;
    if (z == LLAB - 1) un = 0.0f;

    ubar[tid] = 2.0f * un - uo;
    u[tid]    = un;

    if (computeNrj) {
        __shared__ float red[256];
        red[threadIdx.x] = fabsf(un - uo);
        __syncthreads();
        for (int s = 128; s > 0; s >>= 1) {
            if (threadIdx.x < s) red[threadIdx.x] += red[threadIdx.x + s];
            __syncthreads();
        }
        if (threadIdx.x == 0) atomicAdd(nrj, red[0]);
    }
}

// ---------------------------------------------------------------------------
__global__ __launch_bounds__(256) void pd_pack(const float* __restrict__ ws,
                                               float* __restrict__ out) {
    int tid = blockIdx.x * 256 + threadIdx.x;
    if (tid < PXL) out[tid] = ws[OFF_U + tid];
    if (tid == 0) {
        float e = ws[OFF_NRJ];
        out[PXL]     = e;
        out[PXL + 1] = e / (float)PXL;    // denom = 256*256*16
        out[PXL + 2] = 0.0f;
    }
}

// ---------------------------------------------------------------------------
extern "C" void kernel_launch(void* const* d_in, const int* in_sizes, int n_in,
                              void* d_out, int out_size, void* d_ws, size_t ws_size,
                              hipStream_t stream) {
    const float* f = (const float*)d_in[0];   // (256,256,1) f32
    float* ws  = (float*)d_ws;
    float* out = (float*)d_out;

    pd_init_u<<<PXL / 256, 256, 0, stream>>>(f, ws);
    pd_zero  <<<2048,      256, 0, stream>>>(ws);

    for (int it = 0; it < 10; ++it) {                 // repeats = 10
        pd_dual  <<<NPIX / 64, 128, 0, stream>>>(f, ws);
        pd_primal<<<PXL / 256, 256, 0, stream>>>(ws, it == 0 ? 1 : 0);
    }

    pd_pack<<<PXL / 256, 256, 0, stream>>>(ws, out);
}